// PPRPowerIteration_68143951118847
// MI455X (gfx1250) — compile-verified
//
#include <hip/hip_runtime.h>
#include <math.h>

typedef float v2f __attribute__((ext_vector_type(2)));
typedef float v8f __attribute__((ext_vector_type(8)));

#define IN_F   512
#define HID    64
#define NCLS   32
#define ALPHA  0.1f
#define NITER  10

// ---------------------------------------------------------------------------
// Kernel 1: fused local = tanh(x @ w1^T) @ w2^T
// One wave (32 lanes) owns a 16-node tile. GEMM1 accumulates 16x64 via
// V_WMMA_F32_16X16X4_F32 over K=512; tanh applied in registers; h tile staged
// through LDS to convert C-layout -> A-layout; GEMM2 produces 16x32.
// ---------------------------------------------------------------------------
__global__ __launch_bounds__(256) void ppr_mlp_wmma(
    const float* __restrict__ x,   // [N, 512]
    const float* __restrict__ w1,  // [64, 512]
    const float* __restrict__ w2,  // [32, 64]
    float* __restrict__ local_out, // [N, 32]
    int n_nodes) {
  __shared__ float lds_h[8][16 * HID];  // 4KB per wave, 32KB per block

  const int wave  = threadIdx.x >> 5;
  const int lane  = threadIdx.x & 31;
  const int tile  = blockIdx.x * 8 + wave;
  const int row0  = tile * 16;
  if (row0 >= n_nodes) return;

  const int lhalf = lane >> 4;   // 0: lanes 0-15 (K+0/K+1), 1: lanes 16-31 (K+2/K+3)
  const int l     = lane & 15;

  // ---- GEMM1: 16x512 @ 512x64 -> 16x64, four 16-wide hidden tiles ----
  v8f cc0 = {}, cc1 = {}, cc2 = {}, cc3 = {};
  const int  xr   = min(row0 + l, n_nodes - 1);                 // clamp for safety
  const float* xrow = x + (size_t)xr * IN_F + lhalf * 2;        // A: row l, K pair
  const float* w1row = w1 + (size_t)l * IN_F + lhalf * 2;       // B: col l, K pair

  for (int k = 0; k < IN_F; k += 4) {
    v2f a  = *(const v2f*)(xrow + k);
    v2f b0 = *(const v2f*)(w1row + k + 0 * 16 * IN_F);
    v2f b1 = *(const v2f*)(w1row + k + 1 * 16 * IN_F);
    v2f b2 = *(const v2f*)(w1row + k + 2 * 16 * IN_F);
    v2f b3 = *(const v2f*)(w1row + k + 3 * 16 * IN_F);
    cc0 = __builtin_amdgcn_wmma_f32_16x16x4_f32(false, a, false, b0, (short)0, cc0, false, false);
    cc1 = __builtin_amdgcn_wmma_f32_16x16x4_f32(false, a, false, b1, (short)0, cc1, false, false);
    cc2 = __builtin_amdgcn_wmma_f32_16x16x4_f32(false, a, false, b2, (short)0, cc2, false, false);
    cc3 = __builtin_amdgcn_wmma_f32_16x16x4_f32(false, a, false, b3, (short)0, cc3, false, false);
  }

  // ---- tanh + C-layout -> LDS row-major h[16][64] ----
  float* hbuf = lds_h[wave];
  #pragma unroll
  for (int v = 0; v < 8; ++v) {
    const int m = v + lhalf * 8;            // C/D: lane half selects M or M+8
    hbuf[m * HID +  0 + l] = tanhf(cc0[v]);
    hbuf[m * HID + 16 + l] = tanhf(cc1[v]);
    hbuf[m * HID + 32 + l] = tanhf(cc2[v]);
    hbuf[m * HID + 48 + l] = tanhf(cc3[v]);
  }
  __builtin_amdgcn_wave_barrier();  // keep DS store->load ordering (same wave, in-order DS)

  // ---- GEMM2: 16x64 @ 64x32 -> 16x32, two 16-wide class tiles ----
  v8f d0 = {}, d1 = {};
  const float* w2row = w2 + (size_t)l * HID + lhalf * 2;
  const float* harow = hbuf + (size_t)l * HID + lhalf * 2;
  #pragma unroll
  for (int k = 0; k < HID; k += 4) {
    v2f a  = *(const v2f*)(harow + k);
    v2f b0 = *(const v2f*)(w2row + k + 0 * 16 * HID);
    v2f b1 = *(const v2f*)(w2row + k + 1 * 16 * HID);
    d0 = __builtin_amdgcn_wmma_f32_16x16x4_f32(false, a, false, b0, (short)0, d0, false, false);
    d1 = __builtin_amdgcn_wmma_f32_16x16x4_f32(false, a, false, b1, (short)0, d1, false, false);
  }

  // ---- store local [16,32] ----
  #pragma unroll
  for (int v = 0; v < 8; ++v) {
    const int r = row0 + v + lhalf * 8;
    if (r < n_nodes) {
      local_out[(size_t)r * NCLS +  0 + l] = d0[v];
      local_out[(size_t)r * NCLS + 16 + l] = d1[v];
    }
  }
}

// ---------------------------------------------------------------------------
// Kernel 2: dst = alpha * local   (float4-vectorized)
// ---------------------------------------------------------------------------
__global__ __launch_bounds__(256) void ppr_init(
    float4* __restrict__ dst, const float4* __restrict__ local, int n4) {
  int i = blockIdx.x * blockDim.x + threadIdx.x;
  if (i < n4) {
    float4 v = local[i];
    dst[i] = make_float4(ALPHA * v.x, ALPHA * v.y, ALPHA * v.z, ALPHA * v.w);
  }
}

// ---------------------------------------------------------------------------
// Kernel 3: dst[rows[e],:] += vals[e] * p[cols[e],:]
// 8 threads per edge, 4 classes each; p row (128B) is L2-resident.
// ---------------------------------------------------------------------------
__global__ __launch_bounds__(256) void ppr_spmm(
    const int* __restrict__ rows, const int* __restrict__ cols,
    const float* __restrict__ vals, const float* __restrict__ p,
    float* __restrict__ dst, int n_edges) {
  int t  = blockIdx.x * blockDim.x + threadIdx.x;
  int e  = t >> 3;
  int cg = (t & 7) * 4;
  if (e >= n_edges) return;
  const int   r = rows[e];
  const int   c = cols[e];
  const float v = vals[e];
  const float4 pv = *(const float4*)(p + (size_t)c * NCLS + cg);
  float* o = dst + (size_t)r * NCLS + cg;
  atomicAdd(o + 0, v * pv.x);
  atomicAdd(o + 1, v * pv.y);
  atomicAdd(o + 2, v * pv.z);
  atomicAdd(o + 3, v * pv.w);
}

// ---------------------------------------------------------------------------
extern "C" void kernel_launch(void* const* d_in, const int* in_sizes, int n_in,
                              void* d_out, int out_size, void* d_ws, size_t ws_size,
                              hipStream_t stream) {
  const float* x    = (const float*)d_in[0];
  const float* w1   = (const float*)d_in[1];
  const float* w2   = (const float*)d_in[2];
  const int*   rows = (const int*)d_in[3];
  const int*   cols = (const int*)d_in[4];
  const float* vals = (const float*)d_in[5];
  float*       out  = (float*)d_out;

  const int n_nodes = in_sizes[0] / IN_F;
  const int n_edges = in_sizes[3];
  const int nelem   = n_nodes * NCLS;

  float* local = (float*)d_ws;                        // [N,32]
  float* pA    = local + (size_t)nelem;               // [N,32]
  float* pB    = pA + (size_t)nelem;                  // [N,32]

  // 1) fused MLP -> local
  const int ntiles = (n_nodes + 15) / 16;
  ppr_mlp_wmma<<<(ntiles + 7) / 8, 256, 0, stream>>>(x, w1, w2, local, n_nodes);

  // 2) 10 power iterations: p <- A_hat p + alpha * local  (start p = local)
  const float* src = local;
  const int n4        = nelem / 4;
  const int initGrid  = (n4 + 255) / 256;
  const int spmmGrid  = (int)(((long long)n_edges * 8 + 255) / 256);
  for (int it = 0; it < NITER; ++it) {
    float* dst = (it == NITER - 1) ? out : ((it & 1) ? pB : pA);
    ppr_init<<<initGrid, 256, 0, stream>>>((float4*)dst, (const float4*)local, n4);
    ppr_spmm<<<spmmGrid, 256, 0, stream>>>(rows, cols, vals, src, dst, n_edges);
    src = dst;
  }
}